// adder2d_30760555774424
// MI455X (gfx1250) — compile-verified
//
#include <hip/hip_runtime.h>
#include <stdint.h>

// ---------------------------------------------------------------------------
// AdderNet 3x3 "conv" (L1 correlation), fp32.
//   out[n,f,oh,ow] = -sum_{c,kh,kw} | x[n,c,oh+kh-1,ow+kw-1] - W[f,c,kh,kw] |
// N=16, C=32, H=W=56, F=64, pad=1, stride=1.
//
// Compute-bound on VALU (L1 distance has no GEMM structure -> WMMA is not
// mathematically applicable). CDNA5-specific paths used:
//   * tensor_load_to_lds (TDM) DMAs the 36KB W-slice into LDS (verified in
//     disasm round 1)
//   * s_wait_tensorcnt fences the TDM op
// Core loop: 2 VALU ops per element (v_sub_f32 + v_add_f32 with |.| input
// modifier), 7 independent accumulator chains/lane for VOPD dual-issue.
//
// Round-2 fix: lane mapping fl = tid>>3 / grp = tid&7 so that within a wave
// the sW row index spans only 4 values (8-lane broadcast, <=2-way bank
// conflict) and sX window addresses are spaced 7 words (conflict-free),
// instead of the previous fl = tid&31 mapping whose 288-word row stride
// (== 32 mod 64 banks) caused a 16-way LDS bank conflict on every W load.
// ---------------------------------------------------------------------------

#define NB   16
#define CI   32
#define HH   56
#define WW   56
#define FF   64
#define KK   288           // CI * 3 * 3
#define FG   32            // filters per block (2 f-groups)
#define SXW  58            // 56 + 2 halo columns
#define PIX  7             // output pixels per thread (8 groups * 7 = 56)

#if defined(__HIP_DEVICE_COMPILE__) && __has_builtin(__builtin_amdgcn_tensor_load_to_lds)
#define HAVE_TDM 1
#else
#define HAVE_TDM 0
#endif

#if defined(__HIP_DEVICE_COMPILE__) && __has_builtin(__builtin_amdgcn_s_wait_tensorcnt)
#define HAVE_TENSORCNT 1
#else
#define HAVE_TENSORCNT 0
#endif

typedef uint32_t u32x4 __attribute__((ext_vector_type(4)));
typedef int      i32x4 __attribute__((ext_vector_type(4)));
typedef int      i32x8 __attribute__((ext_vector_type(8)));

__global__ __launch_bounds__(256)
void adder2d_kernel(const float* __restrict__ x,
                    const float* __restrict__ Wt,
                    float* __restrict__ out) {
    __shared__ __align__(16) float sW[FG * KK];        // 36,864 B
    __shared__ __align__(16) float sX[CI * 3 * SXW];   // 22,272 B

    const int tid = threadIdx.x;
    const int oh  = blockIdx.x;   // 0..55
    const int fg  = blockIdx.y;   // 0..1
    const int n   = blockIdx.z;   // 0..15

    // ---- stage W slice [fg*32 .. fg*32+31][288] into LDS via TDM ----
    const float* wsrc = Wt + (size_t)fg * FG * KK;     // contiguous 9216 floats

#if HAVE_TDM
    if (tid < 32) {   // one TDM issue from wave 0 (TDM ignores EXEC)
        const uint32_t lds_addr = (uint32_t)(uintptr_t)(void*)sW; // flat low 32b == LDS offset
        const uint64_t ga = (uint64_t)(uintptr_t)wsrc;
        const uint32_t TD0 = (uint32_t)(FG * KK);      // 9216 elements, 1-D tile

        // D# group 0: count=1 | lds_addr | global_addr | type=2
        u32x4 g0 = { 1u,
                     lds_addr,
                     (uint32_t)(ga & 0xFFFFFFFFu),
                     (uint32_t)((ga >> 32) & 0x01FFFFFFu) | (2u << 30) };

        // D# group 1: mask=0, data_size=2 (4B), dims/tile/stride for 1-D tile
        i32x8 g1 = { (int)(2u << 16),                             // data_size = 4B
                     (int)((TD0 & 0xFFFFu) << 16),                // tensor_dim0 [63:48]
                     (int)(((TD0 >> 16) & 0xFFFFu) | (1u << 16)), // dim0 hi | tensor_dim1=1
                     (int)((TD0 & 0xFFFFu) << 16),                // tile_dim0 [127:112]
                     0,                                           // tile_dim1=0, tile_dim2=0
                     (int)TD0,                                    // tensor_dim0_stride lo
                     0, 0 };
        i32x4 g2 = { 0, 0, 0, 0 };
        i32x4 g3 = { 0, 0, 0, 0 };
#if defined(__clang_major__) && (__clang_major__ >= 23)
        i32x8 g4 = { 0, 0, 0, 0, 0, 0, 0, 0 };
        __builtin_amdgcn_tensor_load_to_lds(g0, g1, g2, g3, g4, 0);
#else
        __builtin_amdgcn_tensor_load_to_lds(g0, g1, g2, g3, 0);
#endif
    }
#else
    for (int idx = tid; idx < FG * KK; idx += 256)
        sW[idx] = wsrc[idx];
#endif

    // ---- stage padded x slab: [c][r=oh-1..oh+1][col=-1..56] into LDS ----
    const float* xn = x + (size_t)n * CI * HH * WW;
    for (int idx = tid; idx < CI * 3 * SXW; idx += 256) {
        const int c   = idx / (3 * SXW);
        const int rem = idx - c * (3 * SXW);
        const int r   = rem / SXW;
        const int j   = rem - r * SXW;
        const int ih  = oh - 1 + r;
        const int iw  = j - 1;
        float v = 0.0f;
        if ((unsigned)ih < (unsigned)HH && (unsigned)iw < (unsigned)WW)
            v = xn[(c * HH + ih) * WW + iw];
        sX[idx] = v;
    }

#if HAVE_TDM
    if (tid < 32) {
#if HAVE_TENSORCNT
        __builtin_amdgcn_s_wait_tensorcnt(0);
#else
        asm volatile("s_wait_tensorcnt 0x0" ::: "memory");
#endif
    }
#endif
    __syncthreads();

    // ---- compute: thread -> (f_local = tid>>3, 7 consecutive ow by tid&7) ----
    // Within a wave: 4 distinct f rows (8-lane broadcast on sW, <=2-way bank
    // conflict) and 8 distinct x windows spaced 7 words (conflict-free).
    const int fl  = tid >> 3;         // 0..31
    const int grp = tid & 7;          // 0..7
    const int ow0 = grp * PIX;        // 0,7,...,49

    float acc[PIX];
#pragma unroll
    for (int p = 0; p < PIX; ++p) acc[p] = 0.0f;

    const float* __restrict__ wrow = &sW[fl * KK];

    for (int c = 0; c < CI; ++c) {
        // register-stage 3 rows x 9 cols of the x slab (halo already in LDS)
        float xr[3][PIX + 2];
#pragma unroll
        for (int r = 0; r < 3; ++r)
#pragma unroll
            for (int j = 0; j < PIX + 2; ++j)
                xr[r][j] = sX[(c * 3 + r) * SXW + ow0 + j];

        const float* __restrict__ wk = wrow + c * 9;
#pragma unroll
        for (int kh = 0; kh < 3; ++kh)
#pragma unroll
            for (int kw = 0; kw < 3; ++kw) {
                const float wv = wk[kh * 3 + kw];
#pragma unroll
                for (int p = 0; p < PIX; ++p)
                    acc[p] += fabsf(xr[kh][p + kw] - wv);  // v_sub + v_add |.|
            }
    }

    // ---- write out[n][f][oh][ow0..ow0+6] = -acc ----
    const int f = fg * FG + fl;
    float* op = out + (((size_t)n * FF + f) * HH + oh) * WW + ow0;
#pragma unroll
    for (int p = 0; p < PIX; ++p)
        op[p] = -acc[p];
}

extern "C" void kernel_launch(void* const* d_in, const int* in_sizes, int n_in,
                              void* d_out, int out_size, void* d_ws, size_t ws_size,
                              hipStream_t stream) {
    (void)in_sizes; (void)n_in; (void)out_size; (void)d_ws; (void)ws_size;
    const float* x  = (const float*)d_in[0];   // [16,32,56,56]
    const float* Wt = (const float*)d_in[1];   // [64,32,3,3]
    float* out = (float*)d_out;                // [16,64,56,56]

    dim3 grid(HH, FF / FG, NB);  // (oh, f-group, n)
    adder2d_kernel<<<grid, 256, 0, stream>>>(x, Wt, out);
}